// ClusterDown_7928509629157
// MI455X (gfx1250) — compile-verified
//
#include <hip/hip_runtime.h>

typedef __attribute__((ext_vector_type(2))) float v2f;
typedef __attribute__((ext_vector_type(4))) float v4f;
typedef __attribute__((ext_vector_type(8))) float v8f;

#define HW 65536          // H*W = 256*256
#define CH 128            // channels
#define NCOL 48           // 5*8 = 40 outputs padded to 48 (3 WMMA N-tiles)
#define KSTEPS 32         // 128 / 4
#define WAVES_PER_BLOCK 4
#define TILES_PER_WAVE 4
#define PB_FLOATS (3 * KSTEPS * 32 * 2)   // 6144 floats = 24 KB
#define CSTRIDE 50        // padded pixel stride for C spill (bank-conflict-free)

// ---------------------------------------------------------------------------
// Repack W (5,8,128) -> WMMA B-fragment order PB[t][s][lane][j]:
//   n = t*16 + (lane&15)   (output column, zero-padded for n>=40)
//   k = s*4 + (lane>>4)*2 + j  (channel)
// so each lane fetches its 2-VGPR B fragment with ONE b64 load.
// ---------------------------------------------------------------------------
__global__ __launch_bounds__(128) void repack_kernel(
    const float* __restrict__ Wm, const float* __restrict__ bias,
    float* __restrict__ PB, float* __restrict__ PBias)
{
    int u = blockIdx.x * blockDim.x + threadIdx.x;
    if (u < PB_FLOATS) {
        int j    = u & 1;
        int lane = (u >> 1) & 31;
        int s    = (u >> 6) & 31;
        int t    = u >> 11;
        int n    = t * 16 + (lane & 15);
        int k    = s * 4 + (lane >> 4) * 2 + j;
        PB[u] = (n < 40) ? Wm[n * CH + k] : 0.0f;
    }
    if (u < NCOL) {
        PBias[u] = (u < 40) ? bias[u] : 0.0f;
    }
}

// ---------------------------------------------------------------------------
// Main kernel: each wave computes a 16-pixel x 48-col GEMM tile with
// v_wmma_f32_16x16x4_f32, then label-selects 8 columns per pixel, +bias, ReLU.
// ---------------------------------------------------------------------------
__global__ __launch_bounds__(128) void clusterdown_kernel(
    const float* __restrict__ image, const long long* __restrict__ clusters,
    const float* __restrict__ PB, const float* __restrict__ PBias,
    float* __restrict__ out, int numTiles)
{
    __shared__ float ldsB[PB_FLOATS];
    __shared__ float ldsBias[NCOL];
    __shared__ float ldsC[WAVES_PER_BLOCK * 16 * CSTRIDE];

    const int tid = threadIdx.x;

    // Stage B fragments + bias into LDS (coalesced b128 loads).
    {
        const v4f* src = (const v4f*)PB;
        v4f*       dst = (v4f*)ldsB;
        #pragma unroll
        for (int r = 0; r < PB_FLOATS / 4 / 128; ++r)
            dst[tid + r * 128] = src[tid + r * 128];
        if (tid < NCOL) ldsBias[tid] = PBias[tid];
    }
    __syncthreads();

    const int wv    = tid >> 5;
    const int lane  = tid & 31;
    const int lcol  = lane & 15;   // pixel within tile (A rows) / N column
    const int lhalf = lane >> 4;   // K-half selector in A/B fragments
    float* myC = ldsC + wv * (16 * CSTRIDE);

    const int waveGlobal = blockIdx.x * WAVES_PER_BLOCK + wv;

    for (int i = 0; i < TILES_PER_WAVE; ++i) {
        int tile = waveGlobal * TILES_PER_WAVE + i;
        if (tile >= numTiles) break;           // uniform per wave
        int q0  = tile * 16;                   // flat pixel index of tile start
        int bq  = q0 >> 16;                    // batch (HW = 65536)
        int hw0 = q0 & (HW - 1);

        // Label for this lane's pixel (both lane halves read the same 16).
        long long labv = clusters[(size_t)q0 + lcol];

        // A fragment pointer: lanes 0-15 -> channel k, lanes 16-31 -> k+2,
        // 16 contiguous pixels per half => coalesced.
        const float* pA = image + ((size_t)(bq * CH + lhalf * 2)) * HW + hw0 + lcol;

        v8f c0 = {}, c1 = {}, c2 = {};
        #pragma unroll
        for (int s = 0; s < KSTEPS; ++s) {
            v2f a;
            a[0] = pA[0];        // K = 4s + 2*lhalf
            a[1] = pA[HW];       // K = 4s + 2*lhalf + 1
            pA += 4 * HW;

            const v2f* lb = (const v2f*)ldsB + (s * 32 + lane);
            v2f b0 = lb[0];            // N-tile 0
            v2f b1 = lb[32 * 32];      // N-tile 1
            v2f b2 = lb[2 * 32 * 32];  // N-tile 2

            c0 = __builtin_amdgcn_wmma_f32_16x16x4_f32(false, a, false, b0, (short)0, c0, false, false);
            c1 = __builtin_amdgcn_wmma_f32_16x16x4_f32(false, a, false, b1, (short)0, c1, false, false);
            c2 = __builtin_amdgcn_wmma_f32_16x16x4_f32(false, a, false, b2, (short)0, c2, false, false);
        }

        // Spill C tile to LDS: pixel p = lhalf*8 + r, column = t*16 + lcol.
        #pragma unroll
        for (int r = 0; r < 8; ++r) {
            int p = lhalf * 8 + r;
            myC[p * CSTRIDE +      lcol] = c0[r];
            myC[p * CSTRIDE + 16 + lcol] = c1[r];
            myC[p * CSTRIDE + 32 + lcol] = c2[r];
        }
        // Same-wave LDS is in-order; fence compiler + wait LDS pipe.
        __asm__ volatile("s_wait_dscnt 0" ::: "memory");

        // Label-select 8 outputs for pixel lcol; each half does 4 m-planes.
        bool valid = ((unsigned long long)labv) < 5ull;
        int  nbase = valid ? ((int)labv * 8) : 0;
        #pragma unroll
        for (int m2 = 0; m2 < 4; ++m2) {
            int m = m2 * 2 + lhalf;
            float v = 0.0f;
            if (valid) {
                int n = nbase + m;
                v = myC[lcol * CSTRIDE + n] + ldsBias[n];
                v = v > 0.0f ? v : 0.0f;
            }
            out[((size_t)(bq * 8 + m)) * HW + hw0 + lcol] = v;
        }
    }
}

// ---------------------------------------------------------------------------
extern "C" void kernel_launch(void* const* d_in, const int* in_sizes, int n_in,
                              void* d_out, int out_size, void* d_ws, size_t ws_size,
                              hipStream_t stream) {
    const float*     image    = (const float*)d_in[0];
    const long long* clusters = (const long long*)d_in[1];
    const float*     Wm       = (const float*)d_in[2];
    const float*     bias     = (const float*)d_in[3];
    float*           out      = (float*)d_out;

    float* PB    = (float*)d_ws;
    float* PBias = PB + PB_FLOATS;

    repack_kernel<<<(PB_FLOATS + 127) / 128, 128, 0, stream>>>(Wm, bias, PB, PBias);

    const int numTiles = 4 * HW / 16;                         // 16384
    const int grid = numTiles / (WAVES_PER_BLOCK * TILES_PER_WAVE); // 1024
    clusterdown_kernel<<<grid, 128, 0, stream>>>(image, clusters, PB, PBias, out, numTiles);
}